// NeuralSparseHG_78735340470803
// MI455X (gfx1250) — compile-verified
//
#include <hip/hip_runtime.h>
#include <stdint.h>

// Problem constants (from the reference file)
#define NUM_NODES_  10000
#define NUM_EDGES_  8192
#define NNZ_        320000
#define TOPK_       5

#define SEG_        256     // max staged incidences per node in LDS (avg is 32)
#define WPB_        8       // waves per block in the top-k kernel (8 * 32 = 256 thr)
#define SCAN_T      1024
#define SCAN_W      (SCAN_T / 32)   // 32 waves

typedef unsigned long long u64;
typedef unsigned int       u32;

// ---------------------------------------------------------------------------
// CDNA5 async global->LDS helpers (ASYNCcnt path).
// Builtin signatures (confirmed by clang diagnostics in an earlier round):
//   b32: (int addrspace(1)*, int addrspace(3)*, imm int offset, imm int cpol)
//   b64: (v2i addrspace(1)*, v2i addrspace(3)*, imm int offset, imm int cpol)
// ---------------------------------------------------------------------------
typedef int v2i_t __attribute__((vector_size(2 * sizeof(int))));

__device__ __forceinline__ void g2l_async_b32(const int* g, int* l) {
#if defined(__AMDGCN__) && __has_builtin(__builtin_amdgcn_global_load_async_to_lds_b32)
  typedef __attribute__((address_space(1))) int gi1_t;
  typedef __attribute__((address_space(3))) int li3_t;
  __builtin_amdgcn_global_load_async_to_lds_b32(
      (gi1_t*)(uintptr_t)(const void*)g,   // inttoptr into AS1, exact pointee 'int'
      (li3_t*)l,                           // same-pointee AS conversion to AS3
      0, 0);
#else
  *l = *g;
#endif
}

__device__ __forceinline__ void g2l_async_b64(const u64* g, u64* l) {
#if defined(__AMDGCN__) && __has_builtin(__builtin_amdgcn_global_load_async_to_lds_b64)
  typedef __attribute__((address_space(1))) v2i_t gv1_t;
  typedef __attribute__((address_space(3))) v2i_t lv3_t;
  v2i_t* lg = (v2i_t*)(void*)l;            // reinterpret pointee in generic space
  __builtin_amdgcn_global_load_async_to_lds_b64(
      (gv1_t*)(uintptr_t)(const void*)g,   // inttoptr into AS1, exact pointee v2i
      (lv3_t*)lg,                          // same-pointee AS conversion to AS3
      0, 0);
#else
  *l = *g;
#endif
}

__device__ __forceinline__ void wait_async0() {
#if defined(__AMDGCN__)
#if __has_builtin(__builtin_amdgcn_s_wait_asynccnt)
  __builtin_amdgcn_s_wait_asynccnt(0);
#else
  asm volatile("s_wait_asynccnt 0x0" ::: "memory");
#endif
  __builtin_amdgcn_wave_barrier();
  asm volatile("" ::: "memory");
#endif
}

// ---------------------------------------------------------------------------
// K0: zero the per-node incidence counts (d_ws is poisoned, not re-zeroed).
// ---------------------------------------------------------------------------
__global__ void hg_zero_counts(int* __restrict__ cnt) {
  int i = blockIdx.x * blockDim.x + threadIdx.x;
  if (i < NUM_NODES_) cnt[i] = 0;
}

// ---------------------------------------------------------------------------
// K1: gather + sigmoid. Emits score, a sortable (score,index) 64-bit key,
// histogram counts, and prefills both pruned output rows with -1.
// key = (float_bits(sigmoid) << 32) | (0xFFFFFFFF - i): unique, monotone in
// score (sigmoid > 0 => positive float bits order as uints), ties broken by
// smaller original index — identical semantics to the stable lexsort.
// ---------------------------------------------------------------------------
__global__ void hg_score_kernel(const long long* __restrict__ edge,
                                const float*     __restrict__ logits,
                                float*           __restrict__ out_scores,
                                long long*       __restrict__ out0,
                                long long*       __restrict__ out1,
                                u64*             __restrict__ keys,
                                int*             __restrict__ cnt) {
  int i = blockIdx.x * blockDim.x + threadIdx.x;
  if (i >= NNZ_) return;
  int v = (int)edge[i];
  int e = (int)edge[NNZ_ + i];
  float x = logits[(size_t)v * NUM_EDGES_ + (size_t)e];
  float s = 1.0f / (1.0f + __expf(-x));
  out_scores[i] = s;
  keys[i] = ((u64)__float_as_uint(s) << 32) | (u64)(0xFFFFFFFFu - (u32)i);
  atomicAdd(&cnt[v], 1);
  out0[i] = -1ll;   // default: pruned
  out1[i] = -1ll;
}

// ---------------------------------------------------------------------------
// K2: single-workgroup exclusive scan of 10000 counts. The whole histogram
// (40 KB) lives in LDS — fits easily in CDNA5's 320 KB/WGP. Counts are pulled
// in with async global->LDS loads, and reset to 0 to double as the scatter
// cursors for K3. The 1024 per-thread chunk sums are scanned with a two-level
// wave32 shuffle scan (no serial loop).
// ---------------------------------------------------------------------------
__global__ void __launch_bounds__(SCAN_T)
hg_scan_kernel(int* __restrict__ cnt, int* __restrict__ offs) {
  __shared__ int sc[NUM_NODES_];
  __shared__ int wsum[SCAN_W];
  const int tid  = threadIdx.x;
  const int lane = tid & 31;
  const int wv   = tid >> 5;

  // Stage all counts into LDS (async global->LDS, ASYNCcnt).
  for (int j = tid; j < NUM_NODES_; j += SCAN_T)
    g2l_async_b32(&cnt[j], &sc[j]);
  wait_async0();
  __syncthreads();

  // Per-thread chunk sum (10 counts per thread).
  const int CH = (NUM_NODES_ + SCAN_T - 1) / SCAN_T;   // 10
  const int base = tid * CH;
  int sum = 0;
  for (int k = 0; k < CH; ++k) {
    int j = base + k;
    if (j < NUM_NODES_) sum += sc[j];
  }

  // Level 1: wave32 inclusive scan of chunk sums.
  int inc = sum;
  for (int d = 1; d < 32; d <<= 1) {
    int y = __shfl_up(inc, d, 32);
    if (lane >= d) inc += y;
  }
  if (lane == 31) wsum[wv] = inc;           // wave totals
  __syncthreads();

  // Level 2: wave 0 scans the 32 wave totals.
  if (wv == 0) {
    int w = wsum[lane];
    for (int d = 1; d < 32; d <<= 1) {
      int y = __shfl_up(w, d, 32);
      if (lane >= d) w += y;
    }
    wsum[lane] = w;                          // inclusive wave-prefix sums
  }
  __syncthreads();

  const int wave_prefix = (wv == 0) ? 0 : wsum[wv - 1];
  const int total       = wsum[SCAN_W - 1];
  int run = wave_prefix + (inc - sum);       // exclusive prefix for this chunk

  // Emit per-node offsets; reset counts to 0 (scatter cursor).
  for (int k = 0; k < CH; ++k) {
    int j = base + k;
    if (j < NUM_NODES_) {
      int v = sc[j];
      offs[j] = run;
      run += v;
      cnt[j] = 0;
    }
  }
  if (tid == 0) offs[NUM_NODES_] = total;
}

// ---------------------------------------------------------------------------
// K3: bucket-partition the keys into per-node contiguous segments.
// ---------------------------------------------------------------------------
__global__ void hg_scatter_kernel(const long long* __restrict__ edge,
                                  const u64*       __restrict__ keys,
                                  const int*       __restrict__ offs,
                                  int*             __restrict__ cursor,
                                  u64*             __restrict__ perm) {
  int i = blockIdx.x * blockDim.x + threadIdx.x;
  if (i >= NNZ_) return;
  int v = (int)edge[i];
  int pos = offs[v] + atomicAdd(&cursor[v], 1);
  perm[pos] = keys[i];
}

// ---------------------------------------------------------------------------
// K4: one wave32 per node. Stage the node's key segment into LDS via async
// b64 loads, then 5 rounds of "max key strictly below threshold" with a
// __shfl_xor butterfly. Keys are unique, so the descending threshold fully
// encodes the already-consumed winners. Lane 0 un-masks each winner in the
// output (overwriting the -1 prefill with the real int64 edge values).
// ---------------------------------------------------------------------------
__global__ void __launch_bounds__(WPB_ * 32)
hg_topk_kernel(const u64*       __restrict__ perm,
               const int*       __restrict__ offs,
               const long long* __restrict__ edge,
               long long*       __restrict__ out0,
               long long*       __restrict__ out1) {
  __shared__ u64 stage[WPB_ * SEG_];
  const int wave = threadIdx.x >> 5;
  const int lane = threadIdx.x & 31;
  const int node = blockIdx.x * WPB_ + wave;
  if (node >= NUM_NODES_) return;           // node is wave-uniform; no barriers below

  const int s0  = offs[node];
  const int s1  = offs[node + 1];
  const int len = s1 - s0;
  if (len <= 0) return;

  u64* buf = &stage[wave * SEG_];
  const bool inLds = (len <= SEG_);
  if (inLds) {
    for (int i = lane; i < len; i += 32)
      g2l_async_b64(&perm[s0 + i], &buf[i]);
    wait_async0();                          // wave-private segment: wait suffices
  } else {
    __builtin_prefetch(&perm[s0 + lane], 0, 0);   // global_prefetch_b8
  }

  u64 thr = ~0ull;
  for (int r = 0; r < TOPK_; ++r) {
    u64 lm = 0;
    for (int i = lane; i < len; i += 32) {
      u64 k = inLds ? buf[i] : perm[s0 + i];
      if (k < thr && k > lm) lm = k;
    }
    for (int m = 16; m >= 1; m >>= 1) {     // wave32 butterfly max
      u64 o = __shfl_xor(lm, m, 32);
      if (o > lm) lm = o;
    }
    if ((lm >> 32) == 0ull) break;          // fewer than TOP_K incidences: keep all
    thr = lm;
    if (lane == 0) {
      u32 idx = 0xFFFFFFFFu - (u32)(lm & 0xFFFFFFFFull);
      out0[idx] = edge[idx];                // un-mask the kept incidence
      out1[idx] = edge[NNZ_ + idx];
    }
  }
}

// ---------------------------------------------------------------------------
// Host-side launcher.
// Inputs : d_in[0] = edge_index (int64, [2, NNZ]) ; d_in[1] = logits (f32).
// Output : d_out = pruned_edge_index (int64, [2,NNZ]) then scores (f32, [NNZ]),
//          concatenated flat in return order.
// Scratch: keys (u64[NNZ]) | perm (u64[NNZ]) | cnt (i32[N]) | offs (i32[N+1])
//          = ~5.2 MB total.
// ---------------------------------------------------------------------------
extern "C" void kernel_launch(void* const* d_in, const int* in_sizes, int n_in,
                              void* d_out, int out_size, void* d_ws, size_t ws_size,
                              hipStream_t stream) {
  (void)in_sizes; (void)n_in; (void)out_size; (void)ws_size;

  const long long* edge   = (const long long*)d_in[0];
  const float*     logits = (const float*)d_in[1];

  long long* out0       = (long long*)d_out;
  long long* out1       = out0 + NNZ_;
  float*     out_scores = (float*)(out0 + 2 * NNZ_);

  u64* keys = (u64*)d_ws;
  u64* perm = keys + NNZ_;
  int* cnt  = (int*)(perm + NNZ_);
  int* offs = cnt + NUM_NODES_;

  hg_zero_counts   <<<(NUM_NODES_ + 255) / 256, 256, 0, stream>>>(cnt);
  hg_score_kernel  <<<(NNZ_ + 255) / 256,       256, 0, stream>>>(edge, logits, out_scores,
                                                                  out0, out1, keys, cnt);
  hg_scan_kernel   <<<1, SCAN_T, 0, stream>>>(cnt, offs);
  hg_scatter_kernel<<<(NNZ_ + 255) / 256,       256, 0, stream>>>(edge, keys, offs, cnt, perm);
  hg_topk_kernel   <<<(NUM_NODES_ + WPB_ - 1) / WPB_, WPB_ * 32, 0, stream>>>(perm, offs, edge,
                                                                              out0, out1);
}